// CompletionLoss_20822001451047
// MI455X (gfx1250) — compile-verified
//
#include <hip/hip_runtime.h>
#include <math.h>

// CDNA5 (gfx1250) wave32 WMMA fragments for V_WMMA_F32_16X16X4_F32:
//   A (16x4 f32)  = 2 VGPRs/lane : lanes 0-15 hold K={0,1} for M=lane%16, lanes 16-31 hold K={2,3}
//   B (4x16 f32)  = 2 VGPRs/lane : lanes 0-15 hold K={0,1} for N=lane%16, lanes 16-31 hold K={2,3}
//   C/D (16x16)   = 8 VGPRs/lane : element (M = v + 8*(lane>=16), N = lane%16) in vgpr v
//
// Trick: dist^2(a,b) = ||a||^2 + (||b||^2 - 2 a.b). We compute the parenthesized
// part entirely inside the WMMA by using the spare K=3 slot:
//   A[m,:] = (-2ax, -2ay, -2az, 1)      B[:,n] = (bx, by, bz, ||b_n||^2)
// ||a||^2 is row-constant, so it is added once AFTER the min reduction
// (min is translation-equivariant). Inner loop = ds_load_b64 + wmma + v_min only.
typedef float v2f __attribute__((ext_vector_type(2)));
typedef float v8f __attribute__((ext_vector_type(8)));

#define WAVES_PER_BLOCK 8
#define BLOCK_THREADS   (WAVES_PER_BLOCK * 32)   // 256 threads = 8 wave32
#define ROWS_PER_WAVE   32                        // two 16-row A fragments per wave
#define BLOCK_ROWS      (WAVES_PER_BLOCK * ROWS_PER_WAVE)  // 256 rows per workgroup
#define CHUNK_COLS      256                       // column points staged in LDS per step

__global__ __launch_bounds__(BLOCK_THREADS)
void chamfer_rowmin_wmma(const float* __restrict__ rows_pts,   // [B, nrows, 3]
                         const float* __restrict__ cols_pts,   // [B, ncols, 3]
                         int nrows, int ncols, float scale,
                         float* __restrict__ out)
{
    // Padded layout: per column point 4 floats (x, y, z, ||b||^2) -> 16B stride,
    // so the per-tile B fragment is one conflict-free ds_load_b64 per lane.
    __shared__ float lds_pts[CHUNK_COLS * 4];

    const int b    = blockIdx.y;
    const int lane = threadIdx.x & 31;
    const int wave = threadIdx.x >> 5;
    const int m    = lane & 15;
    const bool hi  = lane >= 16;                  // high half-wave -> K = {2,3}
    const int row0 = blockIdx.x * BLOCK_ROWS + wave * ROWS_PER_WAVE;

    const float* rp = rows_pts + (size_t)b * nrows * 3;
    const float* cp = cols_pts + (size_t)b * ncols * 3;

    // ---- Two A fragments (rows row0..row0+15 and row0+16..row0+31), -2 scaled,
    //      K=3 slot set to 1.0 so the WMMA adds 1 * ||b||^2. ----
    v2f afrag0, afrag1;
    {
        const float* p0 = rp + (size_t)(row0 + m) * 3;
        const float* p1 = rp + (size_t)(row0 + 16 + m) * 3;
        afrag0.x = hi ? (-2.0f * p0[2]) : (-2.0f * p0[0]);
        afrag0.y = hi ? 1.0f            : (-2.0f * p0[1]);
        afrag1.x = hi ? (-2.0f * p1[2]) : (-2.0f * p1[0]);
        afrag1.y = hi ? 1.0f            : (-2.0f * p1[1]);
    }

    // ---- ||a||^2 in the C/D layout: vgpr v -> row row0 + v + (hi?8:0) (+16 for strip 1) ----
    float a2v0[8], a2v1[8];
    #pragma unroll
    for (int v = 0; v < 8; ++v) {
        const float* p0 = rp + (size_t)(row0 + v + (hi ? 8 : 0)) * 3;
        const float* p1 = rp + (size_t)(row0 + 16 + v + (hi ? 8 : 0)) * 3;
        a2v0[v] = p0[0] * p0[0] + p0[1] * p0[1] + p0[2] * p0[2];
        a2v1[v] = p1[0] * p1[0] + p1[1] * p1[1] + p1[2] * p1[2];
    }

    v8f rowmin0, rowmin1;
    #pragma unroll
    for (int v = 0; v < 8; ++v) { rowmin0[v] = 3.0e38f; rowmin1[v] = 3.0e38f; }

    const int bbase = m * 4 + (hi ? 2 : 0);       // lane's dword index into a 16B point slot

    for (int c0 = 0; c0 < ncols; c0 += CHUNK_COLS) {
        __syncthreads();
        // Cooperative stage: 256 threads x 1 point; precompute ||b||^2 once here.
        {
            const float* q = cp + (size_t)(c0 + threadIdx.x) * 3;
            const float qx = q[0], qy = q[1], qz = q[2];
            lds_pts[threadIdx.x * 4 + 0] = qx;
            lds_pts[threadIdx.x * 4 + 1] = qy;
            lds_pts[threadIdx.x * 4 + 2] = qz;
            lds_pts[threadIdx.x * 4 + 3] = qx * qx + qy * qy + qz * qz;
        }
        __syncthreads();

        // Hint next chunk toward the caches while we grind (global_prefetch_b8).
        if (c0 + CHUNK_COLS < ncols)
            __builtin_prefetch(cp + (size_t)(c0 + CHUNK_COLS + threadIdx.x) * 3, 0, 1);

        #pragma unroll
        for (int t = 0; t < CHUNK_COLS; t += 16) {
            // One ds_load_b64: lanes 0-15 get (x,y), lanes 16-31 get (z, ||b||^2).
            const v2f bfrag = *(const v2f*)(&lds_pts[t * 4 + bbase]);

            v8f c0r = {};
            c0r = __builtin_amdgcn_wmma_f32_16x16x4_f32(
                      false, afrag0, false, bfrag, (short)0, c0r, false, false);
            v8f c1r = {};
            c1r = __builtin_amdgcn_wmma_f32_16x16x4_f32(
                      false, afrag1, false, bfrag, (short)0, c1r, false, false);

            #pragma unroll
            for (int v = 0; v < 8; ++v) {
                rowmin0[v] = fminf(rowmin0[v], c0r[v]);   // = ||b||^2 - 2 a.b
                rowmin1[v] = fminf(rowmin1[v], c1r[v]);
            }
        }
    }

    // ---- column (lane) min per 16-lane half, then +||a||^2, sqrt, scaled sum ----
    float wsum = 0.0f;
    #pragma unroll
    for (int v = 0; v < 8; ++v) {
        float mn0 = rowmin0[v];
        float mn1 = rowmin1[v];
        #pragma unroll
        for (int off = 8; off >= 1; off >>= 1) {
            mn0 = fminf(mn0, __shfl_xor(mn0, off, 32));
            mn1 = fminf(mn1, __shfl_xor(mn1, off, 32));
        }
        wsum += sqrtf(fmaxf(mn0 + a2v0[v], 0.0f));   // min(sqrt) == sqrt(min)
        wsum += sqrtf(fmaxf(mn1 + a2v1[v], 0.0f));
    }
    if ((lane & 15) == 0)                             // lanes 0 and 16 hold the half-mins
        atomicAdd(out, wsum * scale);
}

__global__ void zero_out_kernel(float* out) { out[0] = 0.0f; }

extern "C" void kernel_launch(void* const* d_in, const int* in_sizes, int n_in,
                              void* d_out, int out_size, void* d_ws, size_t ws_size,
                              hipStream_t stream) {
    const float* fine   = (const float*)d_in[0];   // [4, 8192, 3]
    const float* coarse = (const float*)d_in[1];   // [4, 1024, 3]
    const float* target = (const float*)d_in[2];   // [4, 8192, 3]
    float* out = (float*)d_out;

    const int B  = 4;
    const int NF = 8192;   // fine / target points
    const int NC = 1024;   // coarse points
    const float ALPHA = 0.5f;

    zero_out_kernel<<<1, 1, 0, stream>>>(out);

    // loss = mean_b[ chamfer(fine,target) + ALPHA * chamfer(coarse,target) ]
    // chamfer(p,t) = mean_i min_j d(p_i,t_j) + mean_j min_i d(p_i,t_j)

    chamfer_rowmin_wmma<<<dim3(NF / BLOCK_ROWS, B), BLOCK_THREADS, 0, stream>>>(
        fine, target, NF, NF, 1.0f / ((float)NF * B), out);          // fine -> target
    chamfer_rowmin_wmma<<<dim3(NF / BLOCK_ROWS, B), BLOCK_THREADS, 0, stream>>>(
        target, fine, NF, NF, 1.0f / ((float)NF * B), out);          // target -> fine
    chamfer_rowmin_wmma<<<dim3(NC / BLOCK_ROWS, B), BLOCK_THREADS, 0, stream>>>(
        coarse, target, NC, NF, ALPHA / ((float)NC * B), out);       // coarse -> target
    chamfer_rowmin_wmma<<<dim3(NF / BLOCK_ROWS, B), BLOCK_THREADS, 0, stream>>>(
        target, coarse, NF, NC, ALPHA / ((float)NF * B), out);       // target -> coarse
}